// AutoRegressiveDecoderLayer_38113539785484
// MI455X (gfx1250) — compile-verified
//
#include <hip/hip_runtime.h>
#include <math.h>

// ---------------------------------------------------------------------------
// MI455X (gfx1250) decoder-layer kernel.
// Memory-bound workload (~1.6 GB KV streaming, ~1 GFLOP): attention kernels
// are flash-style streaming (wave32 per head); the 128x128 linears use
// v_wmma_f32_16x16x4_f32 (only fp32 WMMA shape on CDNA5).
// ---------------------------------------------------------------------------

typedef __attribute__((ext_vector_type(2))) float v2f;
typedef __attribute__((ext_vector_type(8))) float v8f;

#define DIMV    128
#define NHEAD   8
#define DH      16
#define BSZV    1024
#define NKEYSV  1000
#define TPREVV  511
#define TSAV    512
#define LN_EPSF 1e-5f

__device__ __forceinline__ float wred_sum32(float v) {
#pragma unroll
  for (int off = 16; off > 0; off >>= 1) v += __shfl_xor(v, off, 32);
  return v;
}

// ---------------------------------------------------------------------------
// GEMM: Y(M,128) = X(M,128) @ W(128,128) + bias [+ resid] [relu]
// One wave per 16x16 tile; block = 256 threads = 8 waves = full N=128.
// WMMA f32 16x16x4 fragment layouts per CDNA5 ISA 7.12.2:
//   A 16x4 : lanes 0-15 hold M=lane, {v0,v1}={K0,K1}; lanes 16-31 {K2,K3}
//   B 4x16 : lanes 0-15 hold N=lane, {v0,v1}={K0,K1}; lanes 16-31 {K2,K3}
//   C 16x16: vgpr i -> row (i + 8*half), col = lane&15
// ---------------------------------------------------------------------------
__global__ __launch_bounds__(256)
void gemm128_wmma_kernel(const float* __restrict__ X,
                         const float* __restrict__ W,
                         const float* __restrict__ bias,
                         const float* __restrict__ resid,
                         float* __restrict__ Y,
                         int doRelu)
{
  const int m0   = blockIdx.x * 16;
  const int wave = threadIdx.x >> 5;
  const int lane = threadIdx.x & 31;
  const int n0   = wave * 16;
  const int hh   = lane >> 4;   // half-wave select (K pair / row offset)
  const int ml   = lane & 15;

  v8f acc = {};
  const float* xrow = X + (size_t)(m0 + ml) * DIMV;

#pragma unroll
  for (int k0 = 0; k0 < DIMV; k0 += 4) {
    const int ka = k0 + 2 * hh;
    v2f a, b;
    const float2 av = *(const float2*)(xrow + ka);
    a.x = av.x;
    a.y = av.y;
    b.x = W[(size_t)ka * DIMV + n0 + ml];
    b.y = W[(size_t)(ka + 1) * DIMV + n0 + ml];
    acc = __builtin_amdgcn_wmma_f32_16x16x4_f32(
        /*neg_a=*/false, a, /*neg_b=*/false, b,
        /*c_mod=*/(short)0, acc, /*reuse_a=*/false, /*reuse_b=*/false);
  }

  const int col = n0 + ml;
  const float bv = bias[col];
#pragma unroll
  for (int i = 0; i < 8; ++i) {
    const int row = m0 + i + 8 * hh;
    float v = acc[i] + bv;
    if (resid) v += resid[(size_t)row * DIMV + col];
    if (doRelu) v = fmaxf(v, 0.0f);
    Y[(size_t)row * DIMV + col] = v;
  }
}

// ---------------------------------------------------------------------------
// Flash attention, single query token. One block per batch row, one wave per
// head. Lane owns keys t = t0+lane; online softmax state merged by butterfly.
//   Kc/Vc: (BSZ, Lc, DIM) cache.  Knew/Vnew: (BSZ, DIM) appended row (t==Lc).
//   mask : (BSZ, L) bytes, nonzero => score = -1e9 (matches reference).
// ---------------------------------------------------------------------------
__global__ __launch_bounds__(256)
void flash_attn_kernel(const float* __restrict__ Q,
                       const float* __restrict__ Kc,
                       const float* __restrict__ Vc,
                       const float* __restrict__ Knew,
                       const float* __restrict__ Vnew,
                       const unsigned char* __restrict__ mask,
                       float* __restrict__ O,
                       int Lc, int L)
{
  const int b    = blockIdx.x;
  const int h    = threadIdx.x >> 5;
  const int lane = threadIdx.x & 31;
  const float scale = 0.25f;  // 1/sqrt(DH=16)

  // Broadcast-load this head's 16 query values.
  float q[DH];
  {
    const float4* qp = (const float4*)(Q + (size_t)b * DIMV + h * DH);
#pragma unroll
    for (int i = 0; i < 4; ++i) {
      const float4 t = qp[i];
      q[4*i+0] = t.x; q[4*i+1] = t.y; q[4*i+2] = t.z; q[4*i+3] = t.w;
    }
  }

  float mrun = -3.0e38f, lrun = 0.0f;
  float acc[DH];
#pragma unroll
  for (int i = 0; i < DH; ++i) acc[i] = 0.0f;

  const size_t baseKV = (size_t)b * (size_t)Lc * DIMV + (size_t)h * DH;
  const unsigned char* mrow = mask ? (mask + (size_t)b * L) : nullptr;

  for (int t0 = 0; t0 < L; t0 += 32) {
    const int t = t0 + lane;
    if (t + 32 < Lc) {  // prefetch next chunk of this lane's K/V stream
      __builtin_prefetch(Kc + baseKV + (size_t)(t + 32) * DIMV, 0, 1);
      __builtin_prefetch(Vc + baseKV + (size_t)(t + 32) * DIMV, 0, 1);
    }
    if (t < L) {
      const float4* kp;
      const float4* vp;
      if (t < Lc) {
        kp = (const float4*)(Kc + baseKV + (size_t)t * DIMV);
        vp = (const float4*)(Vc + baseKV + (size_t)t * DIMV);
      } else {  // appended current-token K/V
        kp = (const float4*)(Knew + (size_t)b * DIMV + h * DH);
        vp = (const float4*)(Vnew + (size_t)b * DIMV + h * DH);
      }
      float s = 0.0f;
      float vv[DH];
#pragma unroll
      for (int i = 0; i < 4; ++i) {
        const float4 kq = kp[i];
        const float4 vq = vp[i];
        s = fmaf(q[4*i+0], kq.x, s);
        s = fmaf(q[4*i+1], kq.y, s);
        s = fmaf(q[4*i+2], kq.z, s);
        s = fmaf(q[4*i+3], kq.w, s);
        vv[4*i+0] = vq.x; vv[4*i+1] = vq.y; vv[4*i+2] = vq.z; vv[4*i+3] = vq.w;
      }
      s *= scale;
      if (mrow && mrow[t]) s = -1.0e9f;
      const float nm = fmaxf(mrun, s);
      const float sc = __expf(mrun - nm);
      const float p  = __expf(s - nm);
      lrun = lrun * sc + p;
#pragma unroll
      for (int i = 0; i < DH; ++i) acc[i] = acc[i] * sc + p * vv[i];
      mrun = nm;
    }
  }

  // Cross-lane butterfly merge of (m, l, acc[16]); all lanes converge.
#pragma unroll
  for (int off = 16; off > 0; off >>= 1) {
    const float om = __shfl_xor(mrun, off, 32);
    const float ol = __shfl_xor(lrun, off, 32);
    float oacc[DH];
#pragma unroll
    for (int i = 0; i < DH; ++i) oacc[i] = __shfl_xor(acc[i], off, 32);
    const float nm = fmaxf(mrun, om);
    const float s0 = __expf(mrun - nm);
    const float s1 = __expf(om - nm);
    lrun = lrun * s0 + ol * s1;
#pragma unroll
    for (int i = 0; i < DH; ++i) acc[i] = acc[i] * s0 + oacc[i] * s1;
    mrun = nm;
  }

  if (lane == 0) {
    const float inv = 1.0f / lrun;
    float4* op = (float4*)(O + (size_t)b * DIMV + h * DH);
#pragma unroll
    for (int i = 0; i < 4; ++i) {
      float4 o;
      o.x = acc[4*i+0] * inv; o.y = acc[4*i+1] * inv;
      o.z = acc[4*i+2] * inv; o.w = acc[4*i+3] * inv;
      op[i] = o;
    }
  }
}

// ---------------------------------------------------------------------------
// LayerNorm over last dim (128). One wave per row; 8 rows per block.
// ---------------------------------------------------------------------------
__global__ __launch_bounds__(256)
void ln128_kernel(const float* __restrict__ X,
                  const float* __restrict__ g,
                  const float* __restrict__ be,
                  float* __restrict__ Y, int M)
{
  const int row  = blockIdx.x * 8 + (threadIdx.x >> 5);
  const int lane = threadIdx.x & 31;
  if (row >= M) return;
  const float* x = X + (size_t)row * DIMV;
  float v[4];
  float s = 0.0f;
#pragma unroll
  for (int i = 0; i < 4; ++i) { v[i] = x[lane + 32 * i]; s += v[i]; }
  s = wred_sum32(s);
  const float mean = s * (1.0f / DIMV);
  float sq = 0.0f;
#pragma unroll
  for (int i = 0; i < 4; ++i) { const float d = v[i] - mean; sq += d * d; }
  sq = wred_sum32(sq);
  const float rstd = rsqrtf(sq * (1.0f / DIMV) + LN_EPSF);
  float* y = Y + (size_t)row * DIMV;
#pragma unroll
  for (int i = 0; i < 4; ++i) {
    const int c = lane + 32 * i;
    y[c] = (v[i] - mean) * rstd * g[c] + be[c];
  }
}

// ---------------------------------------------------------------------------
extern "C" void kernel_launch(void* const* d_in, const int* in_sizes, int n_in,
                              void* d_out, int out_size, void* d_ws, size_t ws_size,
                              hipStream_t stream) {
  const float* h_t       = (const float*)d_in[0];
  const float* K_att     = (const float*)d_in[1];
  const float* V_att     = (const float*)d_in[2];
  const float* K_sa_prev = (const float*)d_in[3];
  const float* V_sa_prev = (const float*)d_in[4];
  const unsigned char* mask = (const unsigned char*)d_in[5];
  const float* Wq_sa = (const float*)d_in[6];  const float* bq_sa = (const float*)d_in[7];
  const float* Wk_sa = (const float*)d_in[8];  const float* bk_sa = (const float*)d_in[9];
  const float* Wv_sa = (const float*)d_in[10]; const float* bv_sa = (const float*)d_in[11];
  const float* W0_sa = (const float*)d_in[12]; const float* b0_sa = (const float*)d_in[13];
  const float* Wq_a  = (const float*)d_in[14]; const float* bq_a  = (const float*)d_in[15];
  const float* W0_a  = (const float*)d_in[16]; const float* b0_a  = (const float*)d_in[17];
  const float* W1    = (const float*)d_in[18]; const float* b1    = (const float*)d_in[19];
  const float* W2    = (const float*)d_in[20]; const float* b2    = (const float*)d_in[21];
  const float* g_sa  = (const float*)d_in[22]; const float* be_sa = (const float*)d_in[23];
  const float* g_a   = (const float*)d_in[24]; const float* be_a  = (const float*)d_in[25];
  const float* g_mlp = (const float*)d_in[26]; const float* be_mlp= (const float*)d_in[27];

  float* ws = (float*)d_ws;
  const size_t S = (size_t)BSZV * DIMV;  // 131072 floats per buffer
  float* q_sa  = ws + 0 * S;
  float* k_sa  = ws + 1 * S;
  float* v_sa  = ws + 2 * S;
  float* attn1 = ws + 3 * S;
  float* h1p   = ws + 4 * S;
  float* h1    = ws + 5 * S;
  float* q_a   = ws + 6 * S;
  float* attn2 = ws + 7 * S;
  float* h2p   = ws + 8 * S;
  float* h2    = ws + 9 * S;
  float* z     = ws + 10 * S;
  float* outp  = ws + 11 * S;

  const dim3 gG(BSZV / 16), bG(256);
  const dim3 gA(BSZV), bA(256);
  const dim3 gL(BSZV / 8), bL(256);

  // --- self-attention block ---
  gemm128_wmma_kernel<<<gG, bG, 0, stream>>>(h_t, Wq_sa, bq_sa, nullptr, q_sa, 0);
  gemm128_wmma_kernel<<<gG, bG, 0, stream>>>(h_t, Wk_sa, bk_sa, nullptr, k_sa, 0);
  gemm128_wmma_kernel<<<gG, bG, 0, stream>>>(h_t, Wv_sa, bv_sa, nullptr, v_sa, 0);
  flash_attn_kernel<<<gA, bA, 0, stream>>>(q_sa, K_sa_prev, V_sa_prev, k_sa, v_sa,
                                           nullptr, attn1, TPREVV, TSAV);
  gemm128_wmma_kernel<<<gG, bG, 0, stream>>>(attn1, W0_sa, b0_sa, h_t, h1p, 0);
  ln128_kernel<<<gL, bL, 0, stream>>>(h1p, g_sa, be_sa, h1, BSZV);

  // --- masked cross-attention block ---
  gemm128_wmma_kernel<<<gG, bG, 0, stream>>>(h1, Wq_a, bq_a, nullptr, q_a, 0);
  flash_attn_kernel<<<gA, bA, 0, stream>>>(q_a, K_att, V_att, nullptr, nullptr,
                                           mask, attn2, NKEYSV, NKEYSV);
  gemm128_wmma_kernel<<<gG, bG, 0, stream>>>(attn2, W0_a, b0_a, h1, h2p, 0);
  ln128_kernel<<<gL, bL, 0, stream>>>(h2p, g_a, be_a, h2, BSZV);

  // --- MLP block ---
  gemm128_wmma_kernel<<<gG, bG, 0, stream>>>(h2, W1, b1, nullptr, z, 1);
  gemm128_wmma_kernel<<<gG, bG, 0, stream>>>(z, W2, b2, h2, outp, 0);
  ln128_kernel<<<gL, bL, 0, stream>>>(outp, g_mlp, be_mlp, (float*)d_out, BSZV);
}